// MyLayer_45732811767842
// MI455X (gfx1250) — compile-verified
//
#include <hip/hip_runtime.h>

typedef __attribute__((ext_vector_type(2))) float v2f;
typedef __attribute__((ext_vector_type(8))) float v8f;

#define NN   128
#define FD   16
#define OD   64
#define WAVES 4
#define TILE_B 16

// Kernel 1: S[i,j] = (j>i) ? dot64(kernel[i, fd[j], :], kernel[j, fd[i], :]) : 0
__global__ __launch_bounds__(256) void qf_s_kernel(const float* __restrict__ kern,
                                                   const int* __restrict__ fd,
                                                   float* __restrict__ S) {
    int idx = blockIdx.x * blockDim.x + threadIdx.x;
    if (idx >= NN * NN) return;
    int i = idx >> 7;
    int j = idx & (NN - 1);
    float s = 0.f;
    if (j > i) {
        const float* a = kern + (i * FD + fd[j]) * OD;
        const float* b = kern + (j * FD + fd[i]) * OD;
#pragma unroll 8
        for (int k = 0; k < OD; ++k) s += a[k] * b[k];
    }
    S[idx] = s;
}

// Kernel 2: out[b] = x[b]^T Sm x[b] via WMMA f32 16x16x4.
// One wave handles a 16-row b-tile; acc[n] covers output columns n*16..n*16+15.
__global__ __launch_bounds__(WAVES * 32) void qf_main_kernel(const float* __restrict__ x,
                                                             const float* __restrict__ S,
                                                             float* __restrict__ out,
                                                             int Btot) {
    // B fragments of Sm^T pre-swizzled: smB[(kt*8 + n)*32 + lane] = float2 {
    //   Sm[ic][kk], Sm[ic][kk+1] } with ic = n*16 + (lane&15), kk = kt*4 + 2*(lane>>4)
    __shared__ v2f smB[32 * 8 * 32]; // exactly 64 KB
    const int tid  = threadIdx.x;
    const int lane = tid & 31;
    const int wave = tid >> 5;

    for (int idx = tid; idx < 32 * 8 * 32; idx += WAVES * 32) {
        int kt = idx >> 8;
        int n  = (idx >> 5) & 7;
        int l  = idx & 31;
        int ic = n * 16 + (l & 15);
        int kk = kt * 4 + 2 * (l >> 4);
        v2f v;
        v.x = S[ic * NN + kk];
        v.y = S[ic * NN + kk + 1];
        smB[idx] = v;
    }
    __syncthreads();

    const int tile = blockIdx.x * WAVES + wave;
    const int b0   = tile * TILE_B;
    if (b0 >= Btot) return;

    const int row = lane & 15;  // A-matrix M index for this lane
    const int hi  = lane >> 4;  // K sub-offset selector (0 -> K+0/1, 1 -> K+2/3)

    v8f acc[8];
    v8f z = {0.f, 0.f, 0.f, 0.f, 0.f, 0.f, 0.f, 0.f};
#pragma unroll
    for (int n = 0; n < 8; ++n) acc[n] = z;

    const float* xrow = x + (size_t)(b0 + row) * NN + 2 * hi;
#pragma unroll 4
    for (int kt = 0; kt < 32; ++kt) {
        // A fragment (ISA 16x4 f32 layout): lanes 0-15 K=kt*4+0/1, lanes 16-31 K=kt*4+2/3
        v2f a = *(const v2f*)(xrow + kt * 4);
        const v2f* bp = &smB[(kt * 8) * 32 + lane];
#pragma unroll
        for (int n = 0; n < 8; ++n) {
            v2f bfrag = bp[n * 32];
            acc[n] = __builtin_amdgcn_wmma_f32_16x16x4_f32(
                false, a, false, bfrag, (short)0, acc[n], false, false);
        }
    }

    // Epilogue: out[b] = sum_i x[b,i] * T[b,i]
    // C/D layout: acc[n][r] at lane -> row b0 + r + 8*hi, col n*16 + (lane&15)
    const int col = lane & 15;
#pragma unroll
    for (int r = 0; r < 8; ++r) {
        int brow = b0 + hi * 8 + r;
        const float* xr = x + (size_t)brow * NN + col;
        float s = 0.f;
#pragma unroll
        for (int n = 0; n < 8; ++n) s += xr[n * 16] * acc[n][r];
        // reduce across the 16 lanes of this half-wave (masks < 16 stay in-half)
        s += __shfl_xor(s, 1, 32);
        s += __shfl_xor(s, 2, 32);
        s += __shfl_xor(s, 4, 32);
        s += __shfl_xor(s, 8, 32);
        if (col == 0) out[brow] = s;
    }
}

extern "C" void kernel_launch(void* const* d_in, const int* in_sizes, int n_in,
                              void* d_out, int out_size, void* d_ws, size_t ws_size,
                              hipStream_t stream) {
    const float* x    = (const float*)d_in[0];   // (B, 128) f32
    const float* kern = (const float*)d_in[1];   // (128, 16, 64) f32
    const int*   fd   = (const int*)d_in[2];     // (128,) i32
    float* out = (float*)d_out;                  // (B,) f32
    float* S   = (float*)d_ws;                   // 128*128 f32 = 64 KB scratch

    const int Btot = in_sizes[0] / NN;           // 65536

    qf_s_kernel<<<(NN * NN + 255) / 256, 256, 0, stream>>>(kern, fd, S);

    int tiles  = (Btot + TILE_B - 1) / TILE_B;   // 4096
    int blocks = (tiles + WAVES - 1) / WAVES;    // 1024
    qf_main_kernel<<<blocks, WAVES * 32, 0, stream>>>(x, S, out, Btot);
}